// IrrepMultiHeadAttention_12996571038349
// MI455X (gfx1250) — compile-verified
//
#include <hip/hip_runtime.h>
#include <math.h>
#include <stdint.h>

#define BB 2
#define NN 256
#define KK 16
#define BN (BB*NN)
#define EPSR 1e-6f
#define JIT 0.1f

typedef __attribute__((ext_vector_type(2))) float v2f;
typedef __attribute__((ext_vector_type(8))) float v8f;

// ---------------------------------------------------------------------------
// 16x16x16 fp32 GEMM on one wave32 via 4 chained V_WMMA_F32_16X16X4_F32.
// A, B are 16x16 row-major float tiles (LDS or global). transB reads B^T.
// Fragment layouts per CDNA5 ISA 7.12.2:
//   A 16x4:  lane l<16 holds A[l][k0+0..1], lane l>=16 holds A[l-16][k0+2..3]
//   B 4x16 / C 16x16: col = lane&15, row block selected by lane>>4
// ---------------------------------------------------------------------------
__device__ __forceinline__ v8f wmma16(const float* A, const float* Bm,
                                      bool transB, int lane, v8f c) {
  const int half = lane >> 4;
  const int l16  = lane & 15;
#pragma unroll
  for (int k0 = 0; k0 < 16; k0 += 4) {
    const int ka = k0 + 2 * half;
    v2f a, b;
    a.x = A[l16 * 16 + ka + 0];
    a.y = A[l16 * 16 + ka + 1];
    if (!transB) {
      b.x = Bm[(ka + 0) * 16 + l16];
      b.y = Bm[(ka + 1) * 16 + l16];
    } else {
      b.x = Bm[l16 * 16 + ka + 0];
      b.y = Bm[l16 * 16 + ka + 1];
    }
    c = __builtin_amdgcn_wmma_f32_16x16x4_f32(false, a, false, b,
                                              (short)0, c, false, false);
  }
  return c;
}

// Wave-synchronous in-place 16x16 lower Cholesky (lane-per-row, right-looking).
__device__ __forceinline__ float cholesky16(float* S, int lane) {
  float logdet = 0.0f;
  for (int k = 0; k < 16; ++k) {
    float dkk = sqrtf(fmaxf(S[k * 16 + k], 1e-30f));
    logdet += 2.0f * __logf(dkk);
    float inv = 1.0f / dkk;
    __syncthreads();
    if (lane == k) S[k * 16 + k] = dkk;
    if (lane > k && lane < 16) S[lane * 16 + k] *= inv;
    __syncthreads();
    if (lane > k && lane < 16) {
      float lrk = S[lane * 16 + k];
      for (int c2 = k + 1; c2 <= lane; ++c2)
        S[lane * 16 + c2] -= lrk * S[c2 * 16 + k];
    }
    __syncthreads();
  }
  return logdet;
}

// ---------------------------------------------------------------------------
// CDNA5 async global->LDS copy (ASYNCcnt-tracked). One 16x16 f32 tile = 1KB
// = 2 lane-parallel B128 transfers (32 lanes x 16B each).
// LDS byte address = low 32 bits of the flat address of the __shared__ buffer.
// ---------------------------------------------------------------------------
__device__ __forceinline__ void async_tile_load(float* ldsDst,
                                                const float* g, int lane) {
  uint32_t l0 = (uint32_t)(uintptr_t)(ldsDst + lane * 4);
  const float* g0 = g + lane * 4;
  asm volatile("global_load_async_to_lds_b128 %0, %1, off"
               :: "v"(l0), "v"(g0) : "memory");
  uint32_t l1 = (uint32_t)(uintptr_t)(ldsDst + 128 + lane * 4);
  const float* g1 = g + 128 + lane * 4;
  asm volatile("global_load_async_to_lds_b128 %0, %1, off"
               :: "v"(l1), "v"(g1) : "memory");
}

template <int N>
__device__ __forceinline__ void wait_async() {
#if __has_builtin(__builtin_amdgcn_s_wait_asynccnt)
  __builtin_amdgcn_s_wait_asynccnt((short)N);
#else
  asm volatile("s_wait_asynccnt %0" :: "n"(N) : "memory");
#endif
}

// ---------------------------------------------------------------------------
// Phase 1: per (b,n) -> E = expm(phi.G) (scaling/squaring, all-WMMA),
// F = E^T (= expm(-phi.G), antisymmetric generator), sigma_q (WMMA),
// L_i = chol(sigma_q + eps I), logdet_i.  One wave32 per (b,n).
// ---------------------------------------------------------------------------
__global__ void __launch_bounds__(32)
phase1_kernel(const float* __restrict__ sigma_raw,
              const float* __restrict__ phi,
              const float* __restrict__ gens,
              float* __restrict__ Eg, float* __restrict__ Fg,
              float* __restrict__ Sqg, float* __restrict__ Lig,
              float* __restrict__ ldg) {
  __shared__ float Amat[256];
  __shared__ float Rmat[256];
  __shared__ float Tmat[256];
  const int lane = (int)threadIdx.x;
  const int half = lane >> 4, l16 = lane & 15;
  const int bn = (int)blockIdx.x;

  const float p0 = phi[bn * 3 + 0];
  const float p1 = phi[bn * 3 + 1];
  const float p2 = phi[bn * 3 + 2];
  const float sc = 1.0f / 64.0f;  // 2^-6 scaling for scaling & squaring
#pragma unroll
  for (int e = 0; e < 8; ++e) {
    int idx = lane * 8 + e;
    float v = p0 * gens[idx] + p1 * gens[256 + idx] + p2 * gens[512 + idx];
    Amat[idx] = v * sc;
    Rmat[idx] = ((idx >> 4) == (idx & 15)) ? 1.0f : 0.0f;
  }
  __syncthreads();

  // Horner Taylor-8: R <- I + (A*R)/k, k = 8..1   (8 WMMA matmuls)
  for (int k = 8; k >= 1; --k) {
    v8f c = {};
    c = wmma16(Amat, Rmat, false, lane, c);
    __syncthreads();
    float invk = 1.0f / (float)k;
#pragma unroll
    for (int v = 0; v < 8; ++v) {
      int M = v + 8 * half;
      Rmat[M * 16 + l16] = c[v] * invk + ((M == l16) ? 1.0f : 0.0f);
    }
    __syncthreads();
  }
  // 6 squarings: R <- R*R   (6 WMMA matmuls)
  for (int s = 0; s < 6; ++s) {
    v8f c = {};
    c = wmma16(Rmat, Rmat, false, lane, c);
    __syncthreads();
#pragma unroll
    for (int v = 0; v < 8; ++v) Rmat[(v + 8 * half) * 16 + l16] = c[v];
    __syncthreads();
  }

  float* Eb = Eg + bn * 256;
  float* Fb = Fg + bn * 256;
#pragma unroll
  for (int e = 0; e < 8; ++e) {
    int idx = lane * 8 + e;
    int r = idx >> 4, cc = idx & 15;
    Eb[idx] = Rmat[idx];
    Fb[idx] = Rmat[cc * 16 + r];   // F = E^T (orthogonal E)
  }

  // sigma_q = Sr*Sr^T/16 + jitter*I   (1 WMMA matmul, transB)
  const float* Sr = sigma_raw + bn * 256;
#pragma unroll
  for (int e = 0; e < 8; ++e) {
    int idx = lane * 8 + e;
    Amat[idx] = Sr[idx];
  }
  __syncthreads();
  v8f cq = {};
  cq = wmma16(Amat, Amat, true, lane, cq);
  __syncthreads();
  float* Sqb = Sqg + bn * 256;
#pragma unroll
  for (int v = 0; v < 8; ++v) {
    int M = v + 8 * half;
    float val = cq[v] * (1.0f / 16.0f) + ((M == l16) ? JIT : 0.0f);
    Tmat[M * 16 + l16] = val;
    Sqb[M * 16 + l16] = val;
  }
  __syncthreads();
  if (lane < 16) Tmat[lane * 16 + lane] += EPSR;
  __syncthreads();
  float logdet = cholesky16(Tmat, lane);
  __syncthreads();
  float* Lib = Lig + bn * 256;
#pragma unroll
  for (int e = 0; e < 8; ++e) {
    int idx = lane * 8 + e;
    int r = idx >> 4, cc = idx & 15;
    Lib[idx] = (cc <= r) ? Tmat[idx] : 0.0f;
  }
  if (lane == 0) ldg[bn] = logdet;
}

// ---------------------------------------------------------------------------
// Phase 2: one wave32 per (b,i). Stream j = 0..N-1 with double-buffered
// CDNA5 async global->LDS tile loads (next j's F/Sigma tiles land in LDS
// while the current iteration runs its 12 WMMAs + Cholesky + solves).
// ---------------------------------------------------------------------------
__global__ void __launch_bounds__(32)
phase2_kernel(const float* __restrict__ mu_q,
              const float* __restrict__ Eg, const float* __restrict__ Fg,
              const float* __restrict__ Sqg, const float* __restrict__ Lig,
              const float* __restrict__ ldg,
              float* __restrict__ out) {
  __shared__ float FjB[2][256];
  __shared__ float SjB[2][256];
  __shared__ float Om[256];
  __shared__ float Tt[256];
  __shared__ float Sw[256];
  __shared__ float mj[16];
  __shared__ float mt[16];
  __shared__ float mi[16];
  __shared__ float mag[16];

  const int lane = (int)threadIdx.x;
  const int half = lane >> 4, l16 = lane & 15;
  const int bi = (int)blockIdx.x;
  const int b = bi >> 8;

  // kick off async loads for j=0 while we set up wave state
  async_tile_load(FjB[0], Fg + (size_t)(b * NN) * 256, lane);
  async_tile_load(SjB[0], Sqg + (size_t)(b * NN) * 256, lane);

  // Preload E_i A-fragments into registers (reused for all 256 j's)
  const float* Ei = Eg + bi * 256;
  v2f aE[4];
#pragma unroll
  for (int t = 0; t < 4; ++t) {
    int ka = t * 4 + 2 * half;
    aE[t].x = Ei[l16 * 16 + ka + 0];
    aE[t].y = Ei[l16 * 16 + ka + 1];
  }
  // L_i column for this lane (lane = column index for the Frobenius solve)
  const float* Lib = Lig + bi * 256;
  float liC[16];
#pragma unroll
  for (int r = 0; r < 16; ++r) liC[r] = (lane < 16) ? Lib[r * 16 + lane] : 0.0f;
  if (lane < 16) mi[lane] = mu_q[bi * 16 + lane];
  const float logdet_i = ldg[bi];

  float mmax = -3.0e38f, ssum = 0.0f, accMu = 0.0f;
  v8f accS = {};
  __syncthreads();

  for (int j = 0; j < NN; ++j) {
    const int buf = j & 1;
    const int bj = b * NN + j;
    // issue async loads for j+1 into the other buffer, then wait for the
    // current buffer (ASYNCcnt decrements in order: <=4 ==> first batch done)
    if (j + 1 < NN) {
      async_tile_load(FjB[buf ^ 1], Fg + (size_t)(bj + 1) * 256, lane);
      async_tile_load(SjB[buf ^ 1], Sqg + (size_t)(bj + 1) * 256, lane);
      wait_async<4>();
    } else {
      wait_async<0>();
    }
    if (lane < 16) mj[lane] = mu_q[bj * 16 + lane];
    __syncthreads();
    const float* Fj = FjB[buf];
    const float* Sj = SjB[buf];

    // Omega = E_i * F_j  (A-fragments cached in registers)
    v8f co = {};
#pragma unroll
    for (int t = 0; t < 4; ++t) {
      int ka = t * 4 + 2 * half;
      v2f bf;
      bf.x = Fj[(ka + 0) * 16 + l16];
      bf.y = Fj[(ka + 1) * 16 + l16];
      co = __builtin_amdgcn_wmma_f32_16x16x4_f32(false, aE[t], false, bf,
                                                 (short)0, co, false, false);
    }
#pragma unroll
    for (int v = 0; v < 8; ++v) Om[(v + 8 * half) * 16 + l16] = co[v];
    __syncthreads();

    // mu_t = Omega * mu_j  (lane-per-row)
    if (lane < 16) {
      float s = 0.0f;
#pragma unroll
      for (int cc = 0; cc < 16; ++cc) s += Om[lane * 16 + cc] * mj[cc];
      mt[lane] = s;
    }

    // T = Omega * Sigma_j ; Sigma_t = T * Omega^T
    v8f ct = {};
    ct = wmma16(Om, Sj, false, lane, ct);
    __syncthreads();
#pragma unroll
    for (int v = 0; v < 8; ++v) Tt[(v + 8 * half) * 16 + l16] = ct[v];
    __syncthreads();
    v8f cs = {};
    cs = wmma16(Tt, Om, true, lane, cs);
    __syncthreads();
#pragma unroll
    for (int v = 0; v < 8; ++v) {
      int M = v + 8 * half;
      Sw[M * 16 + l16] = cs[v] + ((M == l16) ? EPSR : 0.0f);
    }
    __syncthreads();

    float logdet_p = cholesky16(Sw, lane);  // Sw now holds Lp (lower)
    __syncthreads();

    // trace(Sigma_t^-1 Sigma_i) = |Lp^-1 L_i|_F^2 : lane-per-column fwd subst
    float part = 0.0f;
    {
      float w[16];
#pragma unroll
      for (int r = 0; r < 16; ++r) {
        float s = liC[r];
#pragma unroll
        for (int m = 0; m < r; ++m) s -= Sw[r * 16 + m] * w[m];
        w[r] = s / Sw[r * 16 + r];
        part += w[r] * w[r];
      }
      if (lane >= 16) part = 0.0f;
    }
#pragma unroll
    for (int off = 16; off >= 1; off >>= 1) part += __shfl_xor(part, off, 32);
    float trace_term = part;

    // quad term: |Lp^-1 (mu_t - mu_i)|^2 (redundant on all lanes)
    float quad = 0.0f;
    {
      float y[16];
#pragma unroll
      for (int r = 0; r < 16; ++r) {
        float s = mt[r] - mi[r];
#pragma unroll
        for (int m = 0; m < r; ++m) s -= Sw[r * 16 + m] * y[m];
        y[r] = s / Sw[r * 16 + r];
        quad += y[r] * y[r];
      }
    }

    float kl = 0.5f * (trace_term + quad - 16.0f + logdet_p - logdet_i);
    kl = fmaxf(kl, 0.0f);

    // online softmax over j with fused mixture accumulation
    float lg = -kl;
    float mnew = fmaxf(mmax, lg);
    float scale = __expf(mmax - mnew);
    float wgt = __expf(lg - mnew);
    mmax = mnew;
    ssum = ssum * scale + wgt;
    if (lane < 16) accMu = accMu * scale + wgt * mt[lane];
#pragma unroll
    for (int v = 0; v < 8; ++v) {
      int M = v + 8 * half;
      float st = cs[v] + mt[M] * mt[l16];  // Sigma_t + mu_t mu_t^T
      accS[v] = accS[v] * scale + wgt * st;
    }
    __syncthreads();
  }

  float inv = 1.0f / ssum;
  if (lane < 16) {
    float m = accMu * inv;
    mag[lane] = m;
    out[bi * 16 + lane] = m;  // mu_agg (B,N,K) first in concatenated output
  }
  __syncthreads();
  float* outS = out + BN * 16 + bi * 256;  // sigma_agg (B,N,K,K)
#pragma unroll
  for (int v = 0; v < 8; ++v) {
    int M = v + 8 * half;
    outS[M * 16 + l16] = accS[v] * inv - mag[M] * mag[l16];
  }
}

// ---------------------------------------------------------------------------
extern "C" void kernel_launch(void* const* d_in, const int* in_sizes, int n_in,
                              void* d_out, int out_size, void* d_ws,
                              size_t ws_size, hipStream_t stream) {
  const float* mu_q      = (const float*)d_in[0];  // (B,N,K)
  const float* sigma_raw = (const float*)d_in[1];  // (B,N,K,K)
  const float* phi       = (const float*)d_in[2];  // (B,N,3)
  const float* gens      = (const float*)d_in[3];  // (3,K,K)
  float* out = (float*)d_out;

  // workspace: E, F, Sq, Li tiles (BN x 256) + logdet_i (BN)
  float* ws = (float*)d_ws;
  float* Eg  = ws;
  float* Fg  = Eg + (size_t)BN * 256;
  float* Sqg = Fg + (size_t)BN * 256;
  float* Lig = Sqg + (size_t)BN * 256;
  float* ldg = Lig + (size_t)BN * 256;
  (void)in_sizes; (void)n_in; (void)out_size; (void)ws_size;

  phase1_kernel<<<dim3(BN), dim3(32), 0, stream>>>(sigma_raw, phi, gens,
                                                   Eg, Fg, Sqg, Lig, ldg);
  phase2_kernel<<<dim3(BN), dim3(32), 0, stream>>>(mu_q, Eg, Fg, Sqg, Lig,
                                                   ldg, out);
}